// DistanceLayer_52020643889255
// MI455X (gfx1250) — compile-verified
//
#include <hip/hip_runtime.h>

typedef __attribute__((ext_vector_type(2))) float v2f;
typedef __attribute__((ext_vector_type(4))) float v4f;
typedef __attribute__((ext_vector_type(8))) float v8f;

#define KD 4
#define BD 4096
#define TD 32
#define DD 128
#define ROWP 132   // padded LDS row stride (floats): 528B % 256B = 16B -> conflict-free

// d_out layout (flat, reference return order, all as f32):
//   pos_dist : [K,B,T,T]  offset 0
//   neg_dist : [K,B,T,T]  offset 16777216
//   pair_mask: [K,B,T,T]  offset 33554432  (0.0 / 1.0)
//   nbr_embed: [K,B,T,D]  offset 50331648
#define OFF_NEG 16777216ull
#define OFF_MSK 33554432ull
#define OFF_NBR 50331648ull

__global__ __launch_bounds__(256) void distance_layer_kernel(
    const float* __restrict__ semb,     // [K,B,D]
    const int*   __restrict__ slabels,  // [K,B]
    const int*   __restrict__ topk,     // [K,B,T]
    float*       __restrict__ out)
{
    __shared__ float lds_nbr[TD * ROWP];   // 32 padded neighbor rows
    __shared__ float lds_anchor[DD];
    __shared__ float lds_nsim[TD];
    __shared__ int   lds_idx[TD];
    __shared__ int   lds_same[TD];

    const int pair = blockIdx.x;          // k*B + b
    const int k    = pair >> 12;
    const int b    = pair & (BD - 1);
    const int tid  = threadIdx.x;

    const float* semb_k = semb + (size_t)k * BD * DD;
    const int*   slab_k = slabels + k * BD;

    // ---- stage 1: neighbor indices + same-label flags; anchor row to LDS ----
    if (tid < TD) {
        int n = topk[(size_t)pair * TD + tid];
        lds_idx[tid]  = n;
        lds_same[tid] = (slab_k[n] == slab_k[b]) ? 1 : 0;
    } else if (tid < 64) {
        int j = tid - 32;                 // 32 float4 slots cover D=128
        v4f a = *(const v4f*)(semb_k + (size_t)b * DD + j * 4);
        *(v4f*)(&lds_anchor[j * 4]) = a;
    }
    __syncthreads();

    // ---- stage 2: gather 32 neighbor rows -> LDS (padded) + nbr_embed out ----
    float* out_nbr = out + OFF_NBR + (size_t)pair * (TD * DD);
    #pragma unroll
    for (int it = 0; it < 4; ++it) {
        int idx = tid + it * 256;         // 1024 float4 slots = 32 rows x 32 f4
        int row = idx >> 5;
        int col = idx & 31;
        int n   = lds_idx[row];
        v4f v = *(const v4f*)(semb_k + (size_t)n * DD + col * 4);
        *(v4f*)(&lds_nbr[row * ROWP + col * 4]) = v;
        *(v4f*)(out_nbr + idx * 4) = v;   // coalesced b128 stores
    }
    __syncthreads();

    // ---- stage 3: nsim via V_WMMA_F32_16X16X4_F32 (waves 0,1; EXEC all-1s) --
    const int wave = tid >> 5;
    const int lane = tid & 31;
    if (wave < 2) {
        const int h    = wave;            // neighbor half: rows h*16 .. h*16+15
        const int m    = lane & 15;       // A: M = lane%16
        const int koff = (lane >> 4) << 1;// A/B: K offset 0 (lanes 0-15) / 2 (16-31)
        const float* arow = &lds_nbr[(h * 16 + m) * ROWP + koff];
        const float* brow = &lds_anchor[koff];
        v8f acc0 = {}; v8f acc1 = {};     // two chains to hide WMMA RAW latency
        #pragma unroll
        for (int c = 0; c < 32; c += 2) { // 32 K=4 chunks cover D=128
            v2f a0, b0, a1, b1;
            a0.x = arow[c * 4 + 0]; a0.y = arow[c * 4 + 1];
            b0.x = brow[c * 4 + 0]; b0.y = brow[c * 4 + 1];
            a1.x = arow[c * 4 + 4]; a1.y = arow[c * 4 + 5];
            b1.x = brow[c * 4 + 4]; b1.y = brow[c * 4 + 5];
            acc0 = __builtin_amdgcn_wmma_f32_16x16x4_f32(
                false, a0, false, b0, (short)0, acc0, false, false);
            acc1 = __builtin_amdgcn_wmma_f32_16x16x4_f32(
                false, a1, false, b1, (short)0, acc1, false, false);
        }
        // B columns are broadcast-identical, so every N column of D holds the
        // dot products. C/D layout: vgpr j = M=j (lanes 0-15) / M=8+j (16-31).
        if (lane == 0) {
            #pragma unroll
            for (int j = 0; j < 8; ++j) lds_nsim[h * 16 + j] = acc0[j] + acc1[j];
        } else if (lane == 16) {
            #pragma unroll
            for (int j = 0; j < 8; ++j) lds_nsim[h * 16 + 8 + j] = acc0[j] + acc1[j];
        }
    }
    __syncthreads();

    // ---- stage 4: emit pos/neg/mask over the 32x32 pair grid (float4) ------
    float* out_pos = out + (size_t)pair * 1024;
    float* out_neg = out + OFF_NEG + (size_t)pair * 1024;
    float* out_msk = out + OFF_MSK + (size_t)pair * 1024;
    {
        int s  = tid >> 3;                // slot tid covers elems 4t..4t+3
        int d0 = (tid & 7) << 2;
        int   ms = lds_same[s];
        float ps = lds_nsim[s];
        v4f pv, nv, mv;
        #pragma unroll
        for (int j = 0; j < 4; ++j) {
            int d  = d0 + j;
            int mk = ms & (lds_same[d] ^ 1);
            pv[j] = mk ? ps           : 0.0f;
            nv[j] = mk ? lds_nsim[d]  : 0.0f;
            mv[j] = mk ? 1.0f         : 0.0f;
        }
        *(v4f*)(out_pos + tid * 4) = pv;
        *(v4f*)(out_neg + tid * 4) = nv;
        *(v4f*)(out_msk + tid * 4) = mv;
    }
}

extern "C" void kernel_launch(void* const* d_in, const int* in_sizes, int n_in,
                              void* d_out, int out_size, void* d_ws, size_t ws_size,
                              hipStream_t stream) {
    (void)in_sizes; (void)n_in; (void)out_size; (void)d_ws; (void)ws_size;
    const float* semb    = (const float*)d_in[0];
    const int*   slabels = (const int*)d_in[1];
    const int*   topk    = (const int*)d_in[2];
    float*       out     = (float*)d_out;
    // One workgroup per (k,b) pair: 4*4096 = 16384 blocks of 256 threads.
    distance_layer_kernel<<<dim3(KD * BD), dim3(256), 0, stream>>>(
        semb, slabels, topk, out);
}